// LinearNet_19628000543148
// MI455X (gfx1250) — compile-verified
//
#include <hip/hip_runtime.h>
#include <hip/hip_bf16.h>
#include <math.h>

// ext-vector types for the f32 WMMA 16x16x4 path:
//   A (16x4 f32)  = 64 elems / 32 lanes = 2 VGPRs -> float2
//   B (4x16 f32)  = 64 elems / 32 lanes = 2 VGPRs -> float2
//   C/D (16x16 f32) = 256 elems / 32 lanes = 8 VGPRs -> float8
typedef __attribute__((ext_vector_type(2))) float v2f;
typedef __attribute__((ext_vector_type(8))) float v8f;

#define SQRT_2PI 2.5066282746310002f
#define N_ITERS 100
#define N_PAD   112   // 7 tiles of 16 rows

__global__ __launch_bounds__(32)
void vae_loss_kernel(const float* __restrict__ x,    const float* __restrict__ u0,
                     const float* __restrict__ repa,
                     const float* __restrict__ W_e1, const float* __restrict__ b_e1,
                     const float* __restrict__ W_e2, const float* __restrict__ b_e2,
                     const float* __restrict__ W_x1, const float* __restrict__ b_x1,
                     const float* __restrict__ W_x2, const float* __restrict__ b_x2,
                     const float* __restrict__ W_n1, const float* __restrict__ b_n1,
                     const float* __restrict__ W_n2, const float* __restrict__ b_n2,
                     float* __restrict__ out)
{
    __shared__ float s_h[50];            // encoder hidden (ReLU)
    __shared__ float s_mv[2];            // mean, variance
    __shared__ float s_z[N_PAD];         // z, padded to 112 (pad rows hold mean)
    __shared__ float s_hx[N_PAD * 16];   // x-path hidden tiles (10 valid cols)
    __shared__ float s_hn[N_PAD * 16];   // n-path hidden tiles (pre-ReLU)
    __shared__ float s_red[32];          // loss reduction

    const int lane = threadIdx.x;        // single wave32

    // ---------------- Phase A: encoder ----------------
    const float e0 = u0[0], e1 = x[0], e2 = x[1], e3 = x[2];
    for (int j = lane; j < 50; j += 32) {
        float a = W_e1[j*4+0]*e0 + W_e1[j*4+1]*e1 +
                  W_e1[j*4+2]*e2 + W_e1[j*4+3]*e3 + b_e1[j];
        s_h[j] = fmaxf(a, 0.0f);
    }
    __syncthreads();

    if (lane < 2) {
        float acc = b_e2[lane];
        for (int j = 0; j < 50; ++j) acc += W_e2[lane*50 + j] * s_h[j];
        s_mv[lane] = acc;
    }
    __syncthreads();

    const float mean = s_mv[0];
    const float var  = s_mv[1];

    // ---------------- Phase B: z = mean + repa*variance ----------------
    for (int i = lane; i < N_PAD; i += 32) {
        float rp = (i < N_ITERS) ? repa[i] : 0.0f;
        float z  = mean + rp * var;
        s_z[i] = z;
        if (i < N_ITERS) out[1 + i] = z;   // second tuple output
    }
    __syncthreads();

    // ---------------- Phase C: batched 1->10 layers via V_WMMA_F32_16X16X4_F32 ----------------
    // B (4x16): row K=0 carries the 10-wide weight column; K=1..3 are zero.
    // B VGPR0, lanes 0-15 hold K=0 / N=lane.
    const int n16 = lane & 15;
    v2f bX, bN;
    bX.x = (lane < 16 && n16 < 10) ? W_x1[n16] : 0.0f;  bX.y = 0.0f;
    bN.x = (lane < 16 && n16 < 10) ? W_n1[n16] : 0.0f;  bN.y = 0.0f;

    // C = bias broadcast over rows: C[m][n] = b[n] for every VGPR r (all M).
    const float biasX = (n16 < 10) ? b_x1[n16] : 0.0f;
    const float biasN = (n16 < 10) ? b_n1[n16] : 0.0f;
    v8f cX, cN;
#pragma unroll
    for (int r = 0; r < 8; ++r) { cX[r] = biasX; cN[r] = biasN; }

    const int mbase = (lane < 16) ? 0 : 8;   // C/D layout: M = r + 8*(lane>=16)

    for (int t = 0; t < 7; ++t) {
        // A (16x4): column K=0 = z tile; lanes 0-15 VGPR0 hold (M=lane, K=0).
        v2f a;
        a.x = (lane < 16) ? s_z[t*16 + lane] : 0.0f;  // lanes 16-31 hold K=2/3 -> 0
        a.y = 0.0f;                                    // K=1 / K=3 -> 0

        v8f dX = __builtin_amdgcn_wmma_f32_16x16x4_f32(
            false, a, false, bX, (short)0, cX, false, false);
        v8f dN = __builtin_amdgcn_wmma_f32_16x16x4_f32(
            false, a, false, bN, (short)0, cN, false, false);

#pragma unroll
        for (int r = 0; r < 8; ++r) {
            s_hx[(t*16 + mbase + r)*16 + n16] = dX[r];
            s_hn[(t*16 + mbase + r)*16 + n16] = dN[r];
        }
    }
    __syncthreads();

    // ---------------- Phase D: per-row tail + loss ----------------
    const int   nsel   = (int)u0[0] - 1;                 // PARAMS1 row select
    const float ud     = (nsel == 0) ? -1.0f : -4.0f;
    const float sigma  = sqrtf(-1.0f / (2.0f * ud));
    const float normal = 1.0f / (sigma * SQRT_2PI);
    const float x0 = x[0], x1v = x[1], x2v = x[2];
    const float inv_l1den = 1.0f / (SQRT_2PI * fabsf(var));

    float acc = 0.0f;
    for (int i = lane; i < N_ITERS; i += 32) {
        float l2a0 = b_x2[0], l2a1 = b_x2[1], nli = b_n2[0];
#pragma unroll
        for (int c = 0; c < 10; ++c) {
            float hx = s_hx[i*16 + c];
            float hn = fmaxf(s_hn[i*16 + c], 0.0f);   // ReLU on n-path hidden
            l2a0 += W_x2[c]      * hx;
            l2a1 += W_x2[10 + c] * hx;
            nli  += W_n2[c]      * hn;
        }
        float d0 = l2a0 - x0, d1 = l2a1 - x1v, d2 = nli - x2v;
        float nrm = sqrtf(d0*d0 + d1*d1 + d2*d2);

        float z  = s_z[i];
        float rp = repa[i];
        float l1    = expf(z*z*ud) / normal;
        float loss2 = -logf(l1) + nrm * 200.0f;
        float loss1 = expf(-rp*rp*0.5f) * inv_l1den;
        acc += loss2 + logf(loss1);
    }
    s_red[lane] = acc;
    __syncthreads();

    if (lane == 0) {
        float tot = 0.0f;
#pragma unroll
        for (int l = 0; l < 32; ++l) tot += s_red[l];
        out[0] = tot * (1.0f / (float)N_ITERS);
    }
}

extern "C" void kernel_launch(void* const* d_in, const int* in_sizes, int n_in,
                              void* d_out, int out_size, void* d_ws, size_t ws_size,
                              hipStream_t stream)
{
    (void)in_sizes; (void)n_in; (void)out_size; (void)d_ws; (void)ws_size;
    const float* x    = (const float*)d_in[0];
    const float* u0   = (const float*)d_in[1];
    const float* repa = (const float*)d_in[2];
    const float* W_e1 = (const float*)d_in[3];
    const float* b_e1 = (const float*)d_in[4];
    const float* W_e2 = (const float*)d_in[5];
    const float* b_e2 = (const float*)d_in[6];
    const float* W_x1 = (const float*)d_in[7];
    const float* b_x1 = (const float*)d_in[8];
    const float* W_x2 = (const float*)d_in[9];
    const float* b_x2 = (const float*)d_in[10];
    const float* W_n1 = (const float*)d_in[11];
    const float* b_n1 = (const float*)d_in[12];
    const float* W_n2 = (const float*)d_in[13];
    const float* b_n2 = (const float*)d_in[14];
    float* out = (float*)d_out;

    vae_loss_kernel<<<dim3(1), dim3(32), 0, stream>>>(
        x, u0, repa, W_e1, b_e1, W_e2, b_e2,
        W_x1, b_x1, W_x2, b_x2, W_n1, b_n1, W_n2, b_n2, out);
}